// st_gcn_45423574122915
// MI455X (gfx1250) — compile-verified
//
#include <hip/hip_runtime.h>

typedef float v2f __attribute__((ext_vector_type(2)));
typedef float v8f __attribute__((ext_vector_type(8)));

#define Nn   64
#define Cc   64
#define CI   16
#define Tt   300
#define Vv   25
#define Pp   7500          /* T*V */
#define KCc  48
#define NTV  480000.0f
#define EPSf 1e-5f

// ---------------------------------------------------------------------------
// Kernel 1: y1 = x @ w_down^T + b_down  (WMMA f32 16x16x4), + BN1 sum/sumsq
// grid (64, 59) x 256 : 8 waves/block, 1 wave = 16 positions x 16 out-chan
// ---------------------------------------------------------------------------
__global__ __launch_bounds__(256) void k_gemm_down(
    const float* __restrict__ x, const float* __restrict__ w,
    const float* __restrict__ bias, float* __restrict__ y1,
    float* __restrict__ bn1acc /* [0..15]=sum [16..31]=sumsq */) {
  int n = blockIdx.x;
  int wave = threadIdx.x >> 5, lane = threadIdx.x & 31;
  int half = lane >> 4, l16 = lane & 15;
  int pbase = (blockIdx.y * 8 + wave) * 16;
  int p = pbase + l16;
  bool inb = (p < Pp);
  int pc = inb ? p : (Pp - 1);

  // A operand: weights (16 out x 64 in), 16 K-steps of 4, preloaded statically
  v2f a[16];
#pragma unroll
  for (int s = 0; s < 16; ++s) {
    int k0 = 4 * s + 2 * half;
    a[s].x = w[l16 * 64 + k0];
    a[s].y = w[l16 * 64 + k0 + 1];
  }
  v8f acc = {};
  const float* xn = x + (size_t)n * Cc * Pp;
  // FULL unroll: static VGPR operands for every WMMA (no movrels/M0 indexing)
#pragma unroll
  for (int s = 0; s < 16; ++s) {
    int c = 4 * s + 2 * half;
    v2f b;
    b.x = xn[(size_t)c * Pp + pc];
    b.y = xn[(size_t)(c + 1) * Pp + pc];
    acc = __builtin_amdgcn_wmma_f32_16x16x4_f32(false, a[s], false, b,
                                                (short)0, acc, false, false);
  }
  __shared__ float lsum[16], lss[16];
  if (threadIdx.x < 16) { lsum[threadIdx.x] = 0.f; lss[threadIdx.x] = 0.f; }
  __syncthreads();
  float* y1n = y1 + (size_t)n * CI * Pp;
#pragma unroll
  for (int r = 0; r < 8; ++r) {
    int o = r + 8 * half;
    float val = acc[r] + bias[o];
    if (inb) {
      y1n[(size_t)o * Pp + p] = val;
      atomicAdd(&lsum[o], val);
      atomicAdd(&lss[o], val * val);
    }
  }
  __syncthreads();
  if (threadIdx.x < 16) {
    atomicAdd(&bn1acc[threadIdx.x], lsum[threadIdx.x]);
    atomicAdd(&bn1acc[16 + threadIdx.x], lss[threadIdx.x]);
  }
}

// ---------------------------------------------------------------------------
// Generic BN finalize: scale = g*rsqrt(var+eps), shift = bt - mean*scale
// ---------------------------------------------------------------------------
__global__ void k_bn_fin(const float* __restrict__ sum, const float* __restrict__ ss,
                         const float* __restrict__ g, const float* __restrict__ bt,
                         float* __restrict__ scale, float* __restrict__ shift,
                         int Cn, float inv_count) {
  int c = threadIdx.x;
  if (c < Cn) {
    float m = sum[c] * inv_count;
    float v = ss[c] * inv_count - m * m;
    float sc = g[c] * rsqrtf(v + EPSf);
    scale[c] = sc;
    shift[c] = bt[c] - m * sc;
  }
}

// ---------------------------------------------------------------------------
// Kernel 3: y2 = relu(bn1(y1)) @ w_gc^T + b_gc   (48 out = 3 WMMA tiles)
// ---------------------------------------------------------------------------
__global__ __launch_bounds__(256) void k_gemm_gc(
    const float* __restrict__ y1, const float* __restrict__ scale1,
    const float* __restrict__ shift1, const float* __restrict__ w,
    const float* __restrict__ bias, float* __restrict__ y2) {
  int n = blockIdx.x;
  int wave = threadIdx.x >> 5, lane = threadIdx.x & 31;
  int half = lane >> 4, l16 = lane & 15;
  int pbase = (blockIdx.y * 8 + wave) * 16;
  int p = pbase + l16;
  bool inb = (p < Pp);
  int pc = inb ? p : (Pp - 1);

  v2f a[3][4];
#pragma unroll
  for (int gI = 0; gI < 3; ++gI)
#pragma unroll
    for (int s = 0; s < 4; ++s) {
      int k0 = 4 * s + 2 * half;
      int row = gI * 16 + l16;
      a[gI][s].x = w[row * 16 + k0];
      a[gI][s].y = w[row * 16 + k0 + 1];
    }
  v8f acc[3] = {};
  const float* yn = y1 + (size_t)n * CI * Pp;
#pragma unroll
  for (int s = 0; s < 4; ++s) {
    int c0 = 4 * s + 2 * half;
    v2f b;
    b.x = fmaxf(yn[(size_t)c0 * Pp + pc] * scale1[c0] + shift1[c0], 0.f);
    b.y = fmaxf(yn[(size_t)(c0 + 1) * Pp + pc] * scale1[c0 + 1] + shift1[c0 + 1], 0.f);
#pragma unroll
    for (int gI = 0; gI < 3; ++gI)
      acc[gI] = __builtin_amdgcn_wmma_f32_16x16x4_f32(false, a[gI][s], false, b,
                                                      (short)0, acc[gI], false, false);
  }
  if (inb) {
    float* y2n = y2 + (size_t)n * KCc * Pp;
#pragma unroll
    for (int gI = 0; gI < 3; ++gI)
#pragma unroll
      for (int r = 0; r < 8; ++r) {
        int o = gI * 16 + r + 8 * half;
        y2n[(size_t)o * Pp + p] = acc[gI][r] + bias[o];
      }
  }
}

// ---------------------------------------------------------------------------
// Kernel 4: graph contraction  g[n,c,t,w] = sum_{k,v} y2[n,16k+c,t,v]*A[k,v,w]
// + BN2 sum/sumsq.  grid (64,16) x 256
// ---------------------------------------------------------------------------
__global__ __launch_bounds__(256) void k_graph(
    const float* __restrict__ y2, const float* __restrict__ A,
    float* __restrict__ g, float* __restrict__ bn2acc) {
  __shared__ float As[3 * 25 * 25];
  int n = blockIdx.x, c = blockIdx.y;
  for (int i = threadIdx.x; i < 1875; i += 256) As[i] = A[i];
  __syncthreads();
  float lsum = 0.f, lss = 0.f;
  if (threadIdx.x < 250) {
    int wj = threadIdx.x % 25, tr = threadIdx.x / 25;
    for (int tto = 0; tto < 30; ++tto) {
      int t = tto * 10 + tr;
      float accv = 0.f;
#pragma unroll
      for (int k = 0; k < 3; ++k) {
        const float* row = y2 + (((size_t)n * KCc + k * 16 + c) * Tt + t) * Vv;
        const float* Ak = &As[k * 625 + wj];
#pragma unroll
        for (int v = 0; v < 25; ++v) accv += row[v] * Ak[v * 25];
      }
      g[(((size_t)n * CI + c) * Tt + t) * Vv + wj] = accv;
      lsum += accv; lss += accv * accv;
    }
  }
  __shared__ float r1[256], r2[256];
  r1[threadIdx.x] = lsum; r2[threadIdx.x] = lss;
  __syncthreads();
  for (int s = 128; s > 0; s >>= 1) {
    if (threadIdx.x < s) { r1[threadIdx.x] += r1[threadIdx.x + s]; r2[threadIdx.x] += r2[threadIdx.x + s]; }
    __syncthreads();
  }
  if (threadIdx.x == 0) { atomicAdd(&bn2acc[c], r1[0]); atomicAdd(&bn2acc[16 + c], r2[0]); }
}

// ---------------------------------------------------------------------------
// Kernel 6: y3 = relu(bn2(g)) @ w_up^T + b_up  (64 out = 4 WMMA tiles)
// + per-(n,c) sums S1 (SE mean) + per-c sumsq (BN3)
// ---------------------------------------------------------------------------
__global__ __launch_bounds__(256) void k_gemm_up(
    const float* __restrict__ g, const float* __restrict__ scale2,
    const float* __restrict__ shift2, const float* __restrict__ w,
    const float* __restrict__ bias, float* __restrict__ y3,
    float* __restrict__ S1, float* __restrict__ bn3ss) {
  int n = blockIdx.x;
  int wave = threadIdx.x >> 5, lane = threadIdx.x & 31;
  int half = lane >> 4, l16 = lane & 15;
  int pbase = (blockIdx.y * 8 + wave) * 16;
  int p = pbase + l16;
  bool inb = (p < Pp);
  int pc = inb ? p : (Pp - 1);

  v2f a[4][4];
#pragma unroll
  for (int gI = 0; gI < 4; ++gI)
#pragma unroll
    for (int s = 0; s < 4; ++s) {
      int k0 = 4 * s + 2 * half;
      int row = gI * 16 + l16;
      a[gI][s].x = w[row * 16 + k0];
      a[gI][s].y = w[row * 16 + k0 + 1];
    }
  v8f acc[4] = {};
  const float* gn = g + (size_t)n * CI * Pp;
#pragma unroll
  for (int s = 0; s < 4; ++s) {
    int c0 = 4 * s + 2 * half;
    v2f b;
    b.x = fmaxf(gn[(size_t)c0 * Pp + pc] * scale2[c0] + shift2[c0], 0.f);
    b.y = fmaxf(gn[(size_t)(c0 + 1) * Pp + pc] * scale2[c0 + 1] + shift2[c0 + 1], 0.f);
#pragma unroll
    for (int gI = 0; gI < 4; ++gI)
      acc[gI] = __builtin_amdgcn_wmma_f32_16x16x4_f32(false, a[gI][s], false, b,
                                                      (short)0, acc[gI], false, false);
  }
  __shared__ float ls[64], lq[64];
  if (threadIdx.x < 64) { ls[threadIdx.x] = 0.f; lq[threadIdx.x] = 0.f; }
  __syncthreads();
  float* y3n = y3 + (size_t)n * Cc * Pp;
#pragma unroll
  for (int gI = 0; gI < 4; ++gI)
#pragma unroll
    for (int r = 0; r < 8; ++r) {
      int o = gI * 16 + r + 8 * half;
      float val = acc[gI][r] + bias[o];
      if (inb) {
        y3n[(size_t)o * Pp + p] = val;
        atomicAdd(&ls[o], val);
        atomicAdd(&lq[o], val * val);
      }
    }
  __syncthreads();
  if (threadIdx.x < 64) {
    atomicAdd(&S1[n * 64 + threadIdx.x], ls[threadIdx.x]);
    atomicAdd(&bn3ss[threadIdx.x], lq[threadIdx.x]);
  }
}

// ---------------------------------------------------------------------------
// Kernel 7: finalize BN3 + SE MLP (1 block, 64 threads; thread = channel / sample)
// ---------------------------------------------------------------------------
__global__ void k_se(const float* __restrict__ S1, const float* __restrict__ g_up,
                     const float* __restrict__ bt_up, const float* __restrict__ bn3ss,
                     const float* __restrict__ se_w1, const float* __restrict__ se_w2,
                     float* __restrict__ scale3, float* __restrict__ shift3,
                     float* __restrict__ s_se) {
  __shared__ float sc3[64], sh3[64];
  int tid = threadIdx.x;
  float sum = 0.f;
  for (int n = 0; n < 64; ++n) sum += S1[n * 64 + tid];
  float m = sum / NTV;
  float var = bn3ss[tid] / NTV - m * m;
  float sc = g_up[tid] * rsqrtf(var + EPSf);
  sc3[tid] = sc; sh3[tid] = bt_up[tid] - m * sc;
  scale3[tid] = sc; shift3[tid] = sh3[tid];
  __syncthreads();
  int n = tid;
  float hid[16];
#pragma unroll
  for (int h = 0; h < 16; ++h) hid[h] = 0.f;
  for (int c = 0; c < 64; ++c) {
    float sbar = sc3[c] * (S1[n * 64 + c] * (1.f / 7500.f)) + sh3[c];
#pragma unroll
    for (int h = 0; h < 16; ++h) hid[h] += sbar * se_w1[h * 64 + c];
  }
#pragma unroll
  for (int h = 0; h < 16; ++h) hid[h] = fmaxf(hid[h], 0.f);
  for (int c = 0; c < 64; ++c) {
    float o = 0.f;
#pragma unroll
    for (int h = 0; h < 16; ++h) o += hid[h] * se_w2[c * 16 + h];
    s_se[n * 64 + c] = 1.f / (1.f + __expf(-o));
  }
}

// ---------------------------------------------------------------------------
// Kernel 9: h = relu(bn3(y3)*s + x), per-(n,c) sum/sumsq -> T1,T2. grid(64,64)x256
// ---------------------------------------------------------------------------
__global__ __launch_bounds__(256) void k_se_res(
    const float* __restrict__ y3, const float* __restrict__ scale3,
    const float* __restrict__ shift3, const float* __restrict__ s_se,
    const float* __restrict__ x, float* __restrict__ h,
    float* __restrict__ T1, float* __restrict__ T2) {
  int n = blockIdx.x, c = blockIdx.y;
  size_t base = ((size_t)n * 64 + c) * Pp;
  float sc = scale3[c], sh = shift3[c], se = s_se[n * 64 + c];
  float lsum = 0.f, lss = 0.f;
  for (int i = threadIdx.x; i < Pp; i += 256) {
    float v = fmaxf((y3[base + i] * sc + sh) * se + x[base + i], 0.f);
    h[base + i] = v;
    lsum += v; lss += v * v;
  }
  __shared__ float r1[256], r2[256];
  r1[threadIdx.x] = lsum; r2[threadIdx.x] = lss;
  __syncthreads();
  for (int s = 128; s > 0; s >>= 1) {
    if (threadIdx.x < s) { r1[threadIdx.x] += r1[threadIdx.x + s]; r2[threadIdx.x] += r2[threadIdx.x + s]; }
    __syncthreads();
  }
  if (threadIdx.x == 0) { T1[n * 64 + c] = r1[0]; T2[n * 64 + c] = r2[0]; }
}

// ---------------------------------------------------------------------------
// Kernel 10: channel attention MLP + softmax + BN4 scale/shift (1 block, 64 thr)
// ---------------------------------------------------------------------------
__global__ void k_chatt(const float* __restrict__ T1, const float* __restrict__ T2,
                        const float* __restrict__ ca_w1, const float* __restrict__ ca_b1,
                        const float* __restrict__ ca_g1, const float* __restrict__ ca_bt1,
                        const float* __restrict__ ca_w2, const float* __restrict__ ca_b2,
                        const float* __restrict__ ca_g, const float* __restrict__ ca_bt,
                        float* __restrict__ a_ch, float* __restrict__ scale4,
                        float* __restrict__ shift4) {
  __shared__ float q[64][16];
  __shared__ float qs[16], qb[16];
  int n = threadIdx.x;
  for (int j = 0; j < 16; ++j) {
    float acc = ca_b1[j];
    for (int c = 0; c < 64; ++c)
      acc += (T1[n * 64 + c] * (1.f / 7500.f)) * ca_w1[j * 64 + c];
    q[n][j] = acc;
  }
  __syncthreads();
  if (n < 16) {
    float s = 0.f, ss = 0.f;
    for (int i = 0; i < 64; ++i) { float v = q[i][n]; s += v; ss += v * v; }
    float m = s / 64.f, var = ss / 64.f - m * m;
    float sc = ca_g1[n] * rsqrtf(var + EPSf);
    qs[n] = sc; qb[n] = ca_bt1[n] - m * sc;
  }
  __syncthreads();
  float hid[16];
#pragma unroll
  for (int j = 0; j < 16; ++j) hid[j] = fmaxf(q[n][j] * qs[j] + qb[j], 0.f);
  float pre[64];
  float mx = -1e30f;
  for (int c = 0; c < 64; ++c) {
    float acc = ca_b2[c];
#pragma unroll
    for (int j = 0; j < 16; ++j) acc += hid[j] * ca_w2[c * 16 + j];
    pre[c] = acc; mx = fmaxf(mx, acc);
  }
  float den = 0.f;
  for (int c = 0; c < 64; ++c) { pre[c] = __expf(pre[c] - mx); den += pre[c]; }
  for (int c = 0; c < 64; ++c) a_ch[n * 64 + c] = pre[c] / den;
  __syncthreads();
  int c = threadIdx.x;
  float m4 = 0.f, v4 = 0.f;
  for (int i = 0; i < 64; ++i) {
    float a = a_ch[i * 64 + c];
    m4 += a * T1[i * 64 + c];
    v4 += a * a * T2[i * 64 + c];
  }
  m4 /= NTV; v4 = v4 / NTV - m4 * m4;
  float sc = ca_g[c] * rsqrtf(v4 + EPSf);
  scale4[c] = sc; shift4[c] = ca_bt[c] - m4 * sc;
}

// ---------------------------------------------------------------------------
// Kernel 11: h2 = relu(bn4(h*a) + h) + per-(n,c,t) row sums P1/P2 + frame stats
// ---------------------------------------------------------------------------
__global__ __launch_bounds__(256) void k_chapply(
    const float* __restrict__ h, const float* __restrict__ a_ch,
    const float* __restrict__ scale4, const float* __restrict__ shift4,
    float* __restrict__ h2, float* __restrict__ P1, float* __restrict__ P2,
    float* __restrict__ favg, unsigned* __restrict__ fmaxb) {
  int n = blockIdx.x, c = blockIdx.y;
  size_t base = ((size_t)n * 64 + c) * Pp;
  float a = a_ch[n * 64 + c], sc = scale4[c], sh = shift4[c];
  for (int t = threadIdx.x; t < Tt; t += 256) {
    const float* row = h + base + (size_t)t * Vv;
    float* orow = h2 + base + (size_t)t * Vv;
    float rs = 0.f, rq = 0.f, rm = 0.f;
#pragma unroll
    for (int v = 0; v < 25; ++v) {
      float hv = row[v];
      float val = fmaxf((hv * a) * sc + sh + hv, 0.f);
      orow[v] = val;
      rs += val; rq += val * val; rm = fmaxf(rm, val);
    }
    P1[((size_t)n * 64 + c) * Tt + t] = rs;
    P2[((size_t)n * 64 + c) * Tt + t] = rq;
    atomicAdd(&favg[n * Tt + t], rs);
    atomicMax(&fmaxb[n * Tt + t], __float_as_uint(rm));
  }
}

// ---------------------------------------------------------------------------
// Kernel 12: 1D frame conv over T (kernel 9, pad 4)
// ---------------------------------------------------------------------------
__global__ void k_frameconv(const float* __restrict__ favg, const unsigned* __restrict__ fmaxb,
                            const float* __restrict__ fa_w, const float* __restrict__ fa_b,
                            float* __restrict__ att) {
  int n = blockIdx.x, t = threadIdx.x;
  if (t < Tt) {
    float acc = fa_b[0];
#pragma unroll
    for (int i = 0; i < 9; ++i) {
      int tt = t + i - 4;
      if (tt >= 0 && tt < Tt) {
        acc += fa_w[i] * (favg[n * Tt + tt] * (1.f / 1600.f));
        acc += fa_w[9 + i] * __uint_as_float(fmaxb[n * Tt + tt]);
      }
    }
    att[n * Tt + t] = acc;
  }
}

// ---------------------------------------------------------------------------
// Kernel 13: BN5 stats from att, P1, P2   grid 64 (c) x 256
// ---------------------------------------------------------------------------
__global__ __launch_bounds__(256) void k_bn5(
    const float* __restrict__ att, const float* __restrict__ P1,
    const float* __restrict__ P2, const float* __restrict__ fa_g,
    const float* __restrict__ fa_bt, float* __restrict__ scale5,
    float* __restrict__ shift5) {
  int c = blockIdx.x;
  float s = 0.f, q = 0.f;
  for (int i = threadIdx.x; i < Nn * Tt; i += 256) {
    int n = i / Tt, t = i % Tt;
    float a = att[i];
    s += a * P1[((size_t)n * 64 + c) * Tt + t];
    q += a * a * P2[((size_t)n * 64 + c) * Tt + t];
  }
  __shared__ float r1[256], r2[256];
  r1[threadIdx.x] = s; r2[threadIdx.x] = q;
  __syncthreads();
  for (int st = 128; st > 0; st >>= 1) {
    if (threadIdx.x < st) { r1[threadIdx.x] += r1[threadIdx.x + st]; r2[threadIdx.x] += r2[threadIdx.x + st]; }
    __syncthreads();
  }
  if (threadIdx.x == 0) {
    float m = r1[0] / NTV;
    float var = r2[0] / NTV - m * m;
    float sc = fa_g[c] * rsqrtf(var + EPSf);
    scale5[c] = sc; shift5[c] = fa_bt[c] - m * sc;
  }
}

// ---------------------------------------------------------------------------
// Kernel 14: h3 = relu(bn5(h2*att) + h2) + per-(n,c,v) sums Q1/Q2 + joint pv
// ---------------------------------------------------------------------------
__global__ __launch_bounds__(256) void k_frapply(
    const float* __restrict__ h2, const float* __restrict__ att,
    const float* __restrict__ scale5, const float* __restrict__ shift5,
    float* __restrict__ h3, float* __restrict__ Q1, float* __restrict__ Q2,
    float* __restrict__ pv) {
  int n = blockIdx.x, c = blockIdx.y;
  size_t base = ((size_t)n * 64 + c) * Pp;
  float sc = scale5[c], sh = shift5[c];
  int tid = threadIdx.x;
  int v = tid / 10, tc = tid % 10;
  float qs = 0.f, qq = 0.f;
  if (tid < 250) {
    for (int k = 0; k < 30; ++k) {
      int t = tc * 30 + k;
      float hv = h2[base + (size_t)t * Vv + v];
      float a = att[n * Tt + t];
      float val = fmaxf((hv * a) * sc + sh + hv, 0.f);
      h3[base + (size_t)t * Vv + v] = val;
      qs += val; qq += val * val;
    }
  }
  __shared__ float r1[256], r2[256];
  r1[tid] = qs; r2[tid] = qq;
  __syncthreads();
  if (tid < 250 && tc == 0) {
    float s1 = 0.f, s2 = 0.f;
    for (int j = 0; j < 10; ++j) { s1 += r1[tid + j]; s2 += r2[tid + j]; }
    Q1[((size_t)n * 64 + c) * Vv + v] = s1;
    Q2[((size_t)n * 64 + c) * Vv + v] = s2;
    atomicAdd(&pv[n * Vv + v], s1);
  }
}

// ---------------------------------------------------------------------------
// Kernel 15: joint attention MLP + softmax + BN6 (1 block, 64 threads)
// ---------------------------------------------------------------------------
__global__ void k_joint(const float* __restrict__ pv, const float* __restrict__ Q1,
                        const float* __restrict__ Q2, const float* __restrict__ ja_w1,
                        const float* __restrict__ ja_b1, const float* __restrict__ ja_g1,
                        const float* __restrict__ ja_bt1, const float* __restrict__ ja_w2,
                        const float* __restrict__ ja_b2, const float* __restrict__ ja_g,
                        const float* __restrict__ ja_bt, float* __restrict__ aj,
                        float* __restrict__ scale6, float* __restrict__ shift6) {
  __shared__ float q[64][12];
  __shared__ float qs[12], qb[12];
  int n = threadIdx.x;
  for (int j = 0; j < 12; ++j) {
    float acc = ja_b1[j];
    for (int v = 0; v < 25; ++v)
      acc += (pv[n * 25 + v] * (1.f / 19200.f)) * ja_w1[j * 25 + v];
    q[n][j] = acc;
  }
  __syncthreads();
  if (n < 12) {
    float s = 0.f, ss = 0.f;
    for (int i = 0; i < 64; ++i) { float v = q[i][n]; s += v; ss += v * v; }
    float m = s / 64.f, var = ss / 64.f - m * m;
    float sc = ja_g1[n] * rsqrtf(var + EPSf);
    qs[n] = sc; qb[n] = ja_bt1[n] - m * sc;
  }
  __syncthreads();
  float hid[12];
#pragma unroll
  for (int j = 0; j < 12; ++j) hid[j] = fmaxf(q[n][j] * qs[j] + qb[j], 0.f);
  float pre[25];
  float mx = -1e30f;
  for (int v = 0; v < 25; ++v) {
    float acc = ja_b2[v];
#pragma unroll
    for (int j = 0; j < 12; ++j) acc += hid[j] * ja_w2[v * 12 + j];
    pre[v] = acc; mx = fmaxf(mx, acc);
  }
  float den = 0.f;
  for (int v = 0; v < 25; ++v) { pre[v] = __expf(pre[v] - mx); den += pre[v]; }
  for (int v = 0; v < 25; ++v) aj[n * 25 + v] = pre[v] / den;
  __syncthreads();
  int c = threadIdx.x;
  float m6 = 0.f, v6 = 0.f;
  for (int i = 0; i < 64; ++i)
    for (int v = 0; v < 25; ++v) {
      float a = aj[i * 25 + v];
      m6 += a * Q1[((size_t)i * 64 + c) * 25 + v];
      v6 += a * a * Q2[((size_t)i * 64 + c) * 25 + v];
    }
  m6 /= NTV; v6 = v6 / NTV - m6 * m6;
  float sc = ja_g[c] * rsqrtf(v6 + EPSf);
  scale6[c] = sc; shift6[c] = ja_bt[c] - m6 * sc;
}

// ---------------------------------------------------------------------------
// Kernel 16: out = relu(bn6(h3*aj) + h3)
// ---------------------------------------------------------------------------
__global__ __launch_bounds__(256) void k_final(
    const float* __restrict__ h3, const float* __restrict__ aj,
    const float* __restrict__ scale6, const float* __restrict__ shift6,
    float* __restrict__ out) {
  size_t idx = (size_t)blockIdx.x * blockDim.x + threadIdx.x;
  if (idx < (size_t)Nn * Cc * Pp) {
    size_t p = idx % Pp;
    int v = (int)(p % Vv);
    int c = (int)((idx / Pp) % Cc);
    int n = (int)(idx / ((size_t)Pp * Cc));
    float hv = h3[idx];
    out[idx] = fmaxf((hv * aj[n * 25 + v]) * scale6[c] + shift6[c] + hv, 0.f);
  }
}

// ===========================================================================
extern "C" void kernel_launch(void* const* d_in, const int* in_sizes, int n_in,
                              void* d_out, int out_size, void* d_ws, size_t ws_size,
                              hipStream_t stream) {
  (void)in_sizes; (void)n_in; (void)out_size; (void)ws_size;
  const float* x      = (const float*)d_in[0];
  const float* A      = (const float*)d_in[1];
  const float* w_down = (const float*)d_in[2];
  const float* b_down = (const float*)d_in[3];
  const float* g_down = (const float*)d_in[4];
  const float* bt_down= (const float*)d_in[5];
  const float* w_gc   = (const float*)d_in[6];
  const float* b_gc   = (const float*)d_in[7];
  const float* g_gc   = (const float*)d_in[8];
  const float* bt_gc  = (const float*)d_in[9];
  const float* w_up   = (const float*)d_in[10];
  const float* b_up   = (const float*)d_in[11];
  const float* g_up   = (const float*)d_in[12];
  const float* bt_up  = (const float*)d_in[13];
  const float* se_w1  = (const float*)d_in[14];
  const float* se_w2  = (const float*)d_in[15];
  const float* ca_w1  = (const float*)d_in[16];
  const float* ca_b1  = (const float*)d_in[17];
  const float* ca_g1  = (const float*)d_in[18];
  const float* ca_bt1 = (const float*)d_in[19];
  const float* ca_w2  = (const float*)d_in[20];
  const float* ca_b2  = (const float*)d_in[21];
  const float* ca_g   = (const float*)d_in[22];
  const float* ca_bt  = (const float*)d_in[23];
  const float* fa_w   = (const float*)d_in[24];
  const float* fa_b   = (const float*)d_in[25];
  const float* fa_g   = (const float*)d_in[26];
  const float* fa_bt  = (const float*)d_in[27];
  const float* ja_w1  = (const float*)d_in[28];
  const float* ja_b1  = (const float*)d_in[29];
  const float* ja_g1  = (const float*)d_in[30];
  const float* ja_bt1 = (const float*)d_in[31];
  const float* ja_w2  = (const float*)d_in[32];
  const float* ja_b2  = (const float*)d_in[33];
  const float* ja_g   = (const float*)d_in[34];
  const float* ja_bt  = (const float*)d_in[35];

  float* ws = (float*)d_ws;
  float* out = (float*)d_out;   // doubles as h2 scratch mid-pipeline

  // workspace layout (float offsets)
  const size_t o_y1  = 0;                         // 7,680,000  (y1, then graph out)
  const size_t o_big = o_y1  + 7680000;           // 30,720,000 (y3, then h3)
  const size_t o_y2  = o_big + 30720000;          // 23,040,000
  const size_t o_h   = o_y2  + 23040000;          // 30,720,000 (res2)
  const size_t o_P1  = o_h   + 30720000;          // 1,228,800
  const size_t o_P2  = o_P1  + 1228800;           // 1,228,800
  const size_t o_Q1  = o_P2  + 1228800;           // 102,400
  const size_t o_Q2  = o_Q1  + 102400;            // 102,400
  const size_t o_att = o_Q2  + 102400;            // 19,200
  const size_t o_sc1 = o_att + 19200;             // 16+16
  const size_t o_sh1 = o_sc1 + 16;
  const size_t o_sc2 = o_sh1 + 16;
  const size_t o_sh2 = o_sc2 + 16;
  const size_t o_sc3 = o_sh2 + 16;                // 64-wide from here on
  const size_t o_sh3 = o_sc3 + 64;
  const size_t o_sc4 = o_sh3 + 64;
  const size_t o_sh4 = o_sc4 + 64;
  const size_t o_sc5 = o_sh4 + 64;
  const size_t o_sh5 = o_sc5 + 64;
  const size_t o_sc6 = o_sh5 + 64;
  const size_t o_sh6 = o_sc6 + 64;
  const size_t o_sse = o_sh6 + 64;                // 4096
  const size_t o_ach = o_sse + 4096;              // 4096
  const size_t o_aj  = o_ach + 4096;              // 1600
  const size_t o_T1  = o_aj  + 1600;              // 4096
  const size_t o_T2  = o_T1  + 4096;              // 4096
  // atomically-accumulated region (zeroed every launch)
  const size_t o_acc = o_T2  + 4096;
  const size_t o_bn1 = o_acc;                     // 32 (sum16 + ss16)
  const size_t o_bn2 = o_bn1 + 32;                // 32
  const size_t o_b3s = o_bn2 + 32;                // 64 (bn3 sumsq)
  const size_t o_S1  = o_b3s + 64;                // 4096
  const size_t o_fav = o_S1  + 4096;              // 19,200
  const size_t o_fmx = o_fav + 19200;             // 19,200 (uint bits)
  const size_t o_pv  = o_fmx + 19200;             // 1600
  const size_t accN  = (o_pv + 1600) - o_acc;

  hipMemsetAsync((char*)d_ws + o_acc * sizeof(float), 0, accN * sizeof(float), stream);

  dim3 gemmGrid(Nn, 59);    // ceil(469 tiles / 8 waves)

  k_gemm_down<<<gemmGrid, 256, 0, stream>>>(x, w_down, b_down, ws + o_y1, ws + o_bn1);
  k_bn_fin<<<1, 64, 0, stream>>>(ws + o_bn1, ws + o_bn1 + 16, g_down, bt_down,
                                 ws + o_sc1, ws + o_sh1, 16, 1.f / NTV);
  k_gemm_gc<<<gemmGrid, 256, 0, stream>>>(ws + o_y1, ws + o_sc1, ws + o_sh1,
                                          w_gc, b_gc, ws + o_y2);
  k_graph<<<dim3(Nn, CI), 256, 0, stream>>>(ws + o_y2, A, ws + o_y1, ws + o_bn2);
  k_bn_fin<<<1, 64, 0, stream>>>(ws + o_bn2, ws + o_bn2 + 16, g_gc, bt_gc,
                                 ws + o_sc2, ws + o_sh2, 16, 1.f / NTV);
  k_gemm_up<<<gemmGrid, 256, 0, stream>>>(ws + o_y1, ws + o_sc2, ws + o_sh2,
                                          w_up, b_up, ws + o_big, ws + o_S1, ws + o_b3s);
  k_se<<<1, 64, 0, stream>>>(ws + o_S1, g_up, bt_up, ws + o_b3s, se_w1, se_w2,
                             ws + o_sc3, ws + o_sh3, ws + o_sse);
  k_se_res<<<dim3(Nn, Cc), 256, 0, stream>>>(ws + o_big, ws + o_sc3, ws + o_sh3,
                                             ws + o_sse, x, ws + o_h, ws + o_T1, ws + o_T2);
  k_chatt<<<1, 64, 0, stream>>>(ws + o_T1, ws + o_T2, ca_w1, ca_b1, ca_g1, ca_bt1,
                                ca_w2, ca_b2, ca_g, ca_bt,
                                ws + o_ach, ws + o_sc4, ws + o_sh4);
  k_chapply<<<dim3(Nn, Cc), 256, 0, stream>>>(ws + o_h, ws + o_ach, ws + o_sc4, ws + o_sh4,
                                              out, ws + o_P1, ws + o_P2,
                                              ws + o_fav, (unsigned*)(ws + o_fmx));
  k_frameconv<<<Nn, 320, 0, stream>>>(ws + o_fav, (const unsigned*)(ws + o_fmx),
                                      fa_w, fa_b, ws + o_att);
  k_bn5<<<Cc, 256, 0, stream>>>(ws + o_att, ws + o_P1, ws + o_P2, fa_g, fa_bt,
                                ws + o_sc5, ws + o_sh5);
  k_frapply<<<dim3(Nn, Cc), 256, 0, stream>>>(out, ws + o_att, ws + o_sc5, ws + o_sh5,
                                              ws + o_big, ws + o_Q1, ws + o_Q2, ws + o_pv);
  k_joint<<<1, 64, 0, stream>>>(ws + o_pv, ws + o_Q1, ws + o_Q2, ja_w1, ja_b1, ja_g1,
                                ja_bt1, ja_w2, ja_b2, ja_g, ja_bt,
                                ws + o_aj, ws + o_sc6, ws + o_sh6);
  k_final<<<120000, 256, 0, stream>>>(ws + o_big, ws + o_aj, ws + o_sc6, ws + o_sh6, out);
}